// WeightedBP_53704271069588
// MI455X (gfx1250) — compile-verified
//
#include <hip/hip_runtime.h>
#include <math.h>

// ---------------- problem constants (from reference) ----------------
#define N_VN   8000
#define N_CN   4000
#define DV     3
#define DC     6
#define NEDGE  (N_VN * DV)      // 24000
#define BATCH  1250
#define SLAB   256              // batch columns per slab (wave32 aligned)
#define NSLAB  5                // ceil(1250/256)
#define NUM_ITER 5
#define LLR_MAX 20.0f
#define EPSF   1e-12f

// ---------------- CDNA5 async global->LDS path (probe-confirmed) ------------
#if defined(__gfx1250__) && __has_builtin(__builtin_amdgcn_global_load_async_to_lds_b32) && __has_builtin(__builtin_amdgcn_s_wait_asynccnt)
#define HAS_ASYNC_LDS 1
typedef __attribute__((address_space(1))) int* gas_i32p;
typedef __attribute__((address_space(3))) int* las_i32p;
#else
#define HAS_ASYNC_LDS 0
#endif

// ---------------- CDNA5 WMMA wave reduction ---------------------------------
typedef float v2f __attribute__((ext_vector_type(2)));
typedef float v8f __attribute__((ext_vector_type(8)));

#if defined(__gfx1250__) && __has_builtin(__builtin_amdgcn_wmma_f32_16x16x4_f32)
#define HAS_WMMA_F32 1
#else
#define HAS_WMMA_F32 0
#endif

// Sum of the 32 per-lane values of this wave, returned in every lane.
// Uses V_WMMA_F32_16X16X4_F32 with A = ones(16x4): D[i][j] = sum_k B[k][j].
// Putting the 32 values in B's first VGPR (second VGPR = 0) means
// sum_j D[0][j] over the 16 distinct columns == sum of all 32 inputs,
// independent of the exact (k,j)<->lane mapping.
__device__ __forceinline__ float wave_sum32(float v) {
#if HAS_WMMA_F32
    v2f a; a.x = 1.0f; a.y = 1.0f;     // A all-ones (16x4)
    v2f b; b.x = v;    b.y = 0.0f;     // B holds the 32 wave values
    v8f cz = {};
    v8f d = __builtin_amdgcn_wmma_f32_16x16x4_f32(
        false, a, false, b, (short)0, cz, false, false);
    float s = d[0];                    // row-0 column sums (dup'd in hi half)
    s += __shfl_xor(s, 1, 32);
    s += __shfl_xor(s, 2, 32);
    s += __shfl_xor(s, 4, 32);
    s += __shfl_xor(s, 8, 32);
    return s;                          // total of all 32 lane inputs
#else
    float s = v;
    for (int m = 16; m >= 1; m >>= 1) s += __shfl_xor(s, m, 32);
    return s;
#endif
}

__device__ __forceinline__ float sigma2_from(const int* ebno) {
    float eb = (float)ebno[0];
    // no = 1/(10^(eb/10)*0.5); sigma2 = 4/no = 2*10^(eb/10)
    return 2.0f * powf(10.0f, eb * 0.1f);
}

// ---- L slab build: L[v][jp] = s2/2 - sqrt(s2)*noise[j0+jp][v]  (transpose) -
// First function in file so the async->LDS lowering shows in the disasm.
__global__ void k_build_L(const float* __restrict__ noise, const int* __restrict__ ebno,
                          float* __restrict__ Lmat, int j0) {
    __shared__ float tile[32][33];
    int tx = threadIdx.x;            // 0..31
    int ty0 = threadIdx.y;           // 0..7
    int vbase = blockIdx.x * 32;     // N_VN/32 = 250 blocks
    int jbase = j0 + blockIdx.y * 32;// SLAB/32 = 8 blocks
#pragma unroll
    for (int r = 0; r < 4; r++) {
        int ty = ty0 + r * 8;
        int j = jbase + ty;
        int v = vbase + tx;
        if (j < BATCH) {
#if HAS_ASYNC_LDS
            __builtin_amdgcn_global_load_async_to_lds_b32(
                (gas_i32p)(noise + (size_t)j * N_VN + v),
                (las_i32p)&tile[ty][tx],
                0, 0);
#else
            tile[ty][tx] = noise[(size_t)j * N_VN + v];
#endif
        } else {
            tile[ty][tx] = 0.0f;     // padded batch column
        }
    }
#if HAS_ASYNC_LDS
    __builtin_amdgcn_s_wait_asynccnt(0);
#endif
    __syncthreads();
    float s2 = sigma2_from(ebno);
    float sq = sqrtf(s2);
#pragma unroll
    for (int r = 0; r < 4; r++) {
        int ty = ty0 + r * 8;
        int v  = vbase + ty;
        int jp = blockIdx.y * 32 + tx;            // slab-local column
        Lmat[(size_t)v * SLAB + jp] = 0.5f * s2 - sq * tile[tx][ty];
    }
}

// ---- VN update + c_hat + WMMA-accelerated softplus partial -----------------
__global__ void k_vn(const float* __restrict__ msg_cn, const float* __restrict__ Lmat,
                     float* __restrict__ msg_vn, float* __restrict__ chat_out,
                     float* __restrict__ partials, int j0, int write_chat) {
    __shared__ float wsum[SLAB / 32];
    int v  = blockIdx.x;             // N_VN blocks
    int jp = threadIdx.x;
    size_t b = (size_t)v * SLAB + jp;
    float mc0 = msg_cn[b];
    float mc1 = msg_cn[b + (size_t)N_VN * SLAB];
    float mc2 = msg_cn[b + (size_t)2 * N_VN * SLAB];
    float x = Lmat[b] + mc0 + mc1 + mc2;
    msg_vn[b]                         = x - mc0;
    msg_vn[b + (size_t)N_VN * SLAB]   = x - mc1;
    msg_vn[b + (size_t)2*N_VN * SLAB] = x - mc2;

    int j = j0 + jp;
    float z  = -x;                   // c_hat element
    float sp = 0.0f;
    if (j < BATCH) {
        sp = (z > 0.0f) ? (z + log1pf(expf(-z))) : log1pf(expf(z));
        if (write_chat) chat_out[(size_t)j * N_VN + v] = z;
    }
    // reconverged here: EXEC all ones -> WMMA-legal
    float tot = wave_sum32(sp);
    int wid = jp >> 5, lane = jp & 31;
    if (lane == 0) wsum[wid] = tot;
    __syncthreads();
    if (jp == 0) {
        float s = 0.0f;
#pragma unroll
        for (int i = 0; i < SLAB / 32; i++) s += wsum[i];
        partials[blockIdx.x] = s;
    }
}

// ---- outputs c (zeros) and llr = -s2/2 + sqrt(s2)*noise, both (B,n) --------
__global__ void k_llr_c(const float* __restrict__ noise, const int* __restrict__ ebno,
                        float* __restrict__ c_out, float* __restrict__ llr_out) {
    int tid = blockIdx.x * 256 + threadIdx.x;
    if (tid >= BATCH * N_VN) return;
    float s2 = sigma2_from(ebno);
    float sq = sqrtf(s2);
    c_out[tid]   = 0.0f;
    llr_out[tid] = -0.5f * s2 + sq * noise[tid];
}

// ---- build cn -> edge table (degree is exactly DC=6 for every cn) ----------
__global__ void k_zero_counts(int* cnt) {
    int t = blockIdx.x * 256 + threadIdx.x;
    if (t < N_CN) cnt[t] = 0;
}
__global__ void k_scatter(const int* __restrict__ cn_idx, int* __restrict__ cnt,
                          int* __restrict__ cn2e) {
    int e = blockIdx.x * 256 + threadIdx.x;
    if (e >= NEDGE) return;
    int c = cn_idx[e];
    int slot = atomicAdd(&cnt[c], 1);
    cn2e[c * DC + slot] = e;
}
__global__ void k_sort(int* __restrict__ cn2e) {   // deterministic order
    int c = blockIdx.x * 256 + threadIdx.x;
    if (c >= N_CN) return;
    int v[DC];
#pragma unroll
    for (int i = 0; i < DC; i++) v[i] = cn2e[c * DC + i];
#pragma unroll
    for (int i = 1; i < DC; i++) {
        int key = v[i]; int j = i - 1;
        while (j >= 0 && v[j] > key) { v[j + 1] = v[j]; j--; }
        v[j + 1] = key;
    }
#pragma unroll
    for (int i = 0; i < DC; i++) cn2e[c * DC + i] = v[i];
}

// ---- msg_vn init: msg_vn[e] = L[e % n] -------------------------------------
__global__ void k_init_msg(const float* __restrict__ Lmat, float* __restrict__ msg_vn) {
    int e  = blockIdx.x;             // NEDGE blocks
    int jp = threadIdx.x;            // SLAB threads
    int v  = e % N_VN;
    msg_vn[(size_t)e * SLAB + jp] = Lmat[(size_t)v * SLAB + jp];
}

// ---- CN update: fused tanh/log/segment-sum/extrinsic product ---------------
__global__ void k_cn(const float* __restrict__ msg_vn, const float* __restrict__ weights,
                     const int* __restrict__ cn2e, float* __restrict__ msg_cn) {
    int c  = blockIdx.x;             // uniform per block -> scalar gathers
    int jp = threadIdx.x;
    int eidx[DC];
#pragma unroll
    for (int i = 0; i < DC; i++) eidx[i] = cn2e[c * DC + i];
    __builtin_prefetch(msg_vn + (size_t)eidx[0] * SLAB + jp, 0, 1);

    float lt[DC];
    int   negbits = 0, sum_neg = 0;
    float sum_lt = 0.0f;
#pragma unroll
    for (int i = 0; i < DC; i++) {
        int e    = eidx[i];
        float w  = weights[e];
        float mv = msg_vn[(size_t)e * SLAB + jp];
        float mw = fminf(fmaxf(mv * w, -LLR_MAX), LLR_MAX);
        float t  = tanhf(0.5f * mw);
        float l  = logf(fmaxf(fabsf(t), EPSF));
        lt[i] = l; sum_lt += l;
        int ng = (t < 0.0f) ? 1 : 0;
        negbits |= ng << i; sum_neg += ng;
    }
#pragma unroll
    for (int i = 0; i < DC; i++) {
        int ng   = (negbits >> i) & 1;
        int par  = (sum_neg - ng) & 1;
        float sg = 1.0f - 2.0f * (float)par;
        float mag = expf(sum_lt - lt[i]);      // product of other 5 |t|, <= 1
        float prod = sg * mag;
        prod = fminf(fmaxf(prod, -1.0f + 1e-7f), 1.0f - 1e-7f);
        float mc = logf((1.0f + prod) / (1.0f - prod));  // 2*atanh(prod)
        msg_cn[(size_t)eidx[i] * SLAB + jp] = mc;
    }
}

// ---- final deterministic loss reduce ---------------------------------------
__global__ void k_loss(const float* __restrict__ partials, float* __restrict__ out_loss) {
    __shared__ double red[256];
    int t = threadIdx.x;
    double acc = 0.0;
    const int total = NSLAB * NUM_ITER * N_VN;    // 200000
    for (int i = t; i < total; i += 256) acc += (double)partials[i];
    red[t] = acc;
    __syncthreads();
#pragma unroll
    for (int s = 128; s > 0; s >>= 1) {
        if (t < s) red[t] += red[t + s];
        __syncthreads();
    }
    if (t == 0)
        out_loss[0] = (float)(red[0] / ((double)NUM_ITER * (double)BATCH * (double)N_VN));
}

extern "C" void kernel_launch(void* const* d_in, const int* in_sizes, int n_in,
                              void* d_out, int out_size, void* d_ws, size_t ws_size,
                              hipStream_t stream) {
    const float* noise   = (const float*)d_in[0];   // (B, n)
    const float* weights = (const float*)d_in[1];   // (E,)
    // d_in[2] = vn_idx (structure is tile(arange(n),3) -> not needed)
    const int*   cn_idx  = (const int*)d_in[3];     // (E,)
    const int*   ebno    = (const int*)d_in[4];     // scalar

    float* out_c    = (float*)d_out;                          // B*n zeros
    float* out_chat = out_c    + (size_t)BATCH * N_VN;        // B*n
    float* out_llr  = out_chat + (size_t)BATCH * N_VN;        // B*n
    float* out_loss = out_llr  + (size_t)BATCH * N_VN;        // 1

    char* ws = (char*)d_ws;
    float* msg_vn  = (float*)ws;  ws += sizeof(float) * (size_t)NEDGE * SLAB;
    float* msg_cn  = (float*)ws;  ws += sizeof(float) * (size_t)NEDGE * SLAB;
    float* Lmat    = (float*)ws;  ws += sizeof(float) * (size_t)N_VN * SLAB;
    float* partials= (float*)ws;  ws += sizeof(float) * (size_t)NSLAB * NUM_ITER * N_VN;
    int*   cn2e    = (int*)ws;    ws += sizeof(int) * N_CN * DC;
    int*   cnt     = (int*)ws;    ws += sizeof(int) * N_CN;

    // outputs c (zeros) and llr
    {
        int total = BATCH * N_VN;
        k_llr_c<<<(total + 255) / 256, 256, 0, stream>>>(noise, ebno, out_c, out_llr);
    }
    // build per-CN edge lists (deterministic after sort)
    k_zero_counts<<<(N_CN + 255) / 256, 256, 0, stream>>>(cnt);
    k_scatter<<<(NEDGE + 255) / 256, 256, 0, stream>>>(cn_idx, cnt, cn2e);
    k_sort<<<(N_CN + 255) / 256, 256, 0, stream>>>(cn2e);

    // process batch in 5 slabs of 256 columns (keeps workspace ~59 MB)
    for (int s = 0; s < NSLAB; s++) {
        int j0 = s * SLAB;
        dim3 gl(N_VN / 32, SLAB / 32);
        dim3 bl(32, 8);
        k_build_L<<<gl, bl, 0, stream>>>(noise, ebno, Lmat, j0);
        k_init_msg<<<NEDGE, SLAB, 0, stream>>>(Lmat, msg_vn);
        for (int it = 0; it < NUM_ITER; it++) {
            k_cn<<<N_CN, SLAB, 0, stream>>>(msg_vn, weights, cn2e, msg_cn);
            k_vn<<<N_VN, SLAB, 0, stream>>>(msg_cn, Lmat, msg_vn, out_chat,
                                            partials + ((size_t)s * NUM_ITER + it) * N_VN,
                                            j0, (it == NUM_ITER - 1) ? 1 : 0);
        }
    }
    k_loss<<<1, 256, 0, stream>>>(partials, out_loss);
}